// FaceLandmarkEstimator_49555332661966
// MI455X (gfx1250) — compile-verified
//
#include <hip/hip_runtime.h>
#include <math.h>
#include <stdint.h>

// ---------------------------------------------------------------------------
// FaceLandmarkEstimator: per-(b,c) 64x64 argmax + subpixel refine + inv affine
// B=256, C=68, H=W=64.  One wave32 per row; async global->LDS double-buffered
// streaming (CDNA5 GLOBAL_LOAD_ASYNC_TO_LDS_B128 + s_wait_asynccnt).
// ---------------------------------------------------------------------------

#define WAVES_PER_BLOCK 8
#define ROW_FLOATS      4096          // 64*64
#define TILE_FLOATS     512           // 2 KB per tile
#define TILES_PER_ROW   (ROW_FLOATS / TILE_FLOATS)   // 8
#define TILE_BYTES      (TILE_FLOATS * 4)            // 2048

// Issue one 2KB tile as 4 async 512B bursts (32 lanes x 16B each).
// INST_OFFSET applies to both the global address and the LDS address
// (CDNA5 ISA), so one LDS base + one 64-bit vaddr suffices.
__device__ __forceinline__ void issue_tile_async(unsigned lds_byte, const char* gaddr) {
    asm volatile(
        "global_load_async_to_lds_b128 %0, %1, off\n\t"
        "global_load_async_to_lds_b128 %0, %1, off offset:512\n\t"
        "global_load_async_to_lds_b128 %0, %1, off offset:1024\n\t"
        "global_load_async_to_lds_b128 %0, %1, off offset:1536"
        :: "v"(lds_byte), "v"(gaddr)
        : "memory");
}

#define ASYNC_WAIT(N) asm volatile("s_wait_asynccnt %0" :: "n"(N) : "memory")

__global__ __launch_bounds__(WAVES_PER_BLOCK * 32)
void FaceLandmarkEstimator_argmax_kernel(const float* __restrict__ pred_hm,
                                         const float* __restrict__ center,
                                         const float* __restrict__ scale,
                                         float* __restrict__ out,
                                         int nrows)   // B*C
{
    // [buffer][wave][float4 chunk] : 2 * 8 * 128 * 16B = 32 KB
    __shared__ float4 smem[2][WAVES_PER_BLOCK][TILE_FLOATS / 4];

    const int lane = threadIdx.x & 31;
    const int wave = threadIdx.x >> 5;
    const int row  = blockIdx.x * WAVES_PER_BLOCK + wave;   // wave-uniform
    if (row >= nrows) return;

    const float* rowBase = pred_hm + (size_t)row * ROW_FLOATS;

    // LDS byte offsets for this wave's two tile buffers (per-lane 16B slots).
    // Low 32 bits of the generic LDS address == DS byte offset on AMDGPU.
    unsigned lds0 = (unsigned)(uintptr_t)(&smem[0][wave][0]) + (unsigned)lane * 16u;
    unsigned lds1 = (unsigned)(uintptr_t)(&smem[1][wave][0]) + (unsigned)lane * 16u;
    const char* gl = (const char*)rowBase + lane * 16;

    // Prime the double buffer.
    issue_tile_async(lds0, gl);
    issue_tile_async(lds1, gl + TILE_BYTES);

    float bestVal = -INFINITY;
    int   bestIdx = 0;

    for (int t = 0; t < TILES_PER_ROW; ++t) {
        // Async loads complete in order: <=4 outstanding means tile t landed.
        if (t < TILES_PER_ROW - 1) { ASYNC_WAIT(4); } else { ASYNC_WAIT(0); }

        const float4* bufp = &smem[t & 1][wave][0];
#pragma unroll
        for (int k = 0; k < 4; ++k) {
            float4 v = bufp[lane + 32 * k];
            int bi = t * TILE_FLOATS + (lane + 32 * k) * 4;
            // strict '>' keeps the earliest index within this lane's
            // monotonically increasing stream (jnp.argmax first-occurrence).
            if (v.x > bestVal) { bestVal = v.x; bestIdx = bi;     }
            if (v.y > bestVal) { bestVal = v.y; bestIdx = bi + 1; }
            if (v.z > bestVal) { bestVal = v.z; bestIdx = bi + 2; }
            if (v.w > bestVal) { bestVal = v.w; bestIdx = bi + 3; }
        }

        // Refill the buffer we just drained with tile t+2.
        if (t + 2 < TILES_PER_ROW) {
            issue_tile_async((t & 1) ? lds1 : lds0, gl + (t + 2) * TILE_BYTES);
        }
    }

    // Wave32 cross-lane reduction: max value, tie -> smallest flat index.
#pragma unroll
    for (int offs = 16; offs > 0; offs >>= 1) {
        float ov = __shfl_xor(bestVal, offs, 32);
        int   oi = __shfl_xor(bestIdx, offs, 32);
        if (ov > bestVal || (ov == bestVal && oi < bestIdx)) {
            bestVal = ov;
            bestIdx = oi;
        }
    }

    if (lane == 0) {
        const int W = 64, H = 64;
        int x = bestIdx & (W - 1);      // 0-based column
        int y = bestIdx >> 6;           // 0-based row
        int xc = min(max(x, 1), W - 2);
        int yc = min(max(y, 1), H - 2);

        float dx = rowBase[yc * W + xc + 1] - rowBase[yc * W + xc - 1];
        float dy = rowBase[(yc + 1) * W + xc] - rowBase[(yc - 1) * W + xc];
        bool interior = (x > 0) & (x < W - 1) & (y > 0) & (y < H - 1);

        float px = (float)(x + 1);
        float py = (float)(y + 1);
        if (interior) {
            float sx = (float)((dx > 0.0f) - (dx < 0.0f));
            float sy = (float)((dy > 0.0f) - (dy < 0.0f));
            px += 0.25f * sx;
            py += 0.25f * sy;
        }
        px -= 0.5f;
        py -= 0.5f;

        int b = row / 68;               // C == 68
        float h  = 200.0f * scale[b];
        float sH = h * (1.0f / 64.0f);  // h / H, H == 64
        float ox = px * sH + center[2 * b + 0] - 0.5f * h;
        float oy = py * sH + center[2 * b + 1] - 0.5f * h;

        out[2 * row + 0] = ox;
        out[2 * row + 1] = oy;
    }
}

extern "C" void kernel_launch(void* const* d_in, const int* in_sizes, int n_in,
                              void* d_out, int out_size, void* d_ws, size_t ws_size,
                              hipStream_t stream) {
    (void)n_in; (void)out_size; (void)d_ws; (void)ws_size;
    const float* pred_hm = (const float*)d_in[0];
    const float* center  = (const float*)d_in[1];
    const float* scale   = (const float*)d_in[2];
    float*       out     = (float*)d_out;

    int nrows = in_sizes[0] / ROW_FLOATS;   // B*C = 17408
    dim3 block(WAVES_PER_BLOCK * 32);
    dim3 grid((nrows + WAVES_PER_BLOCK - 1) / WAVES_PER_BLOCK);
    FaceLandmarkEstimator_argmax_kernel<<<grid, block, 0, stream>>>(
        pred_hm, center, scale, out, nrows);
}